// CombinedHiddenGCVAE_16286515987222
// MI455X (gfx1250) — compile-verified
//
#include <hip/hip_runtime.h>
#include <math.h>

// Problem dims (from reference)
#define FEAT 128
#define COND 32
#define HID  128
#define LAT  64

typedef float v2f __attribute__((ext_vector_type(2)));
typedef float v8f __attribute__((ext_vector_type(8)));
typedef int   v4i_vs __attribute__((vector_size(16)));   // matches builtin's V4i param type

// gfx1250 async global->LDS path (guarded: falls back to plain copy if builtins absent)
#if defined(__has_builtin)
#if __has_builtin(__builtin_amdgcn_global_load_async_to_lds_b128) && \
    __has_builtin(__builtin_amdgcn_s_wait_asynccnt)
#define ASYNC_LDS_OK 1
#endif
#endif
#ifndef ASYNC_LDS_OK
#define ASYNC_LDS_OK 0
#endif

// ================= CSR construction (degree count -> scan -> fill) =================

__global__ __launch_bounds__(256) void zero_i32(int* __restrict__ p, int n) {
    int i = blockIdx.x * 256 + threadIdx.x;
    if (i < n) p[i] = 0;
}

__global__ __launch_bounds__(256) void count_dst(const int* __restrict__ dst,
                                                 int* __restrict__ cnt, int e) {
    int i = blockIdx.x * 256 + threadIdx.x;
    if (i < e) atomicAdd(&cnt[dst[i]], 1);
}

// dis[i] = rsqrt(deg), deg = 1 (self loop) + in-edge count  (exact integer count)
__global__ __launch_bounds__(256) void dis_k(const int* __restrict__ cnt,
                                             float* __restrict__ dis, int n) {
    int i = blockIdx.x * 256 + threadIdx.x;
    if (i < n) dis[i] = rsqrtf(1.0f + (float)cnt[i]);
}

__global__ __launch_bounds__(256) void scan_block_sums(const int* __restrict__ cnt,
                                                       int* __restrict__ bsum, int n) {
    __shared__ int sm[256];
    const int t = threadIdx.x;
    const int i = blockIdx.x * 256 + t;
    sm[t] = (i < n) ? cnt[i] : 0;
    __syncthreads();
    for (int s = 128; s > 0; s >>= 1) {
        if (t < s) sm[t] += sm[t + s];
        __syncthreads();
    }
    if (t == 0) bsum[blockIdx.x] = sm[0];
}

// single-block exclusive scan of block sums (nb <= 1024)
__global__ __launch_bounds__(1024) void scan_partials(int* __restrict__ bsum, int nb) {
    __shared__ int sm[1024];
    const int t = threadIdx.x;
    sm[t] = (t < nb) ? bsum[t] : 0;
    __syncthreads();
    for (int off = 1; off < 1024; off <<= 1) {
        int v = sm[t];
        if (t >= off) v += sm[t - off];
        __syncthreads();
        sm[t] = v;
        __syncthreads();
    }
    if (t < nb) bsum[t] = (t == 0) ? 0 : sm[t - 1];   // exclusive
}

__global__ __launch_bounds__(256) void scan_final(const int* __restrict__ cnt,
                                                  const int* __restrict__ bsum_ex,
                                                  int* __restrict__ rowstart,
                                                  int* __restrict__ cursor, int n, int E) {
    __shared__ int sm[256];
    const int t = threadIdx.x;
    const int i = blockIdx.x * 256 + t;
    const int v = (i < n) ? cnt[i] : 0;
    sm[t] = v;
    __syncthreads();
    for (int off = 1; off < 256; off <<= 1) {
        int u = sm[t];
        if (t >= off) u += sm[t - off];
        __syncthreads();
        sm[t] = u;
        __syncthreads();
    }
    const int r = bsum_ex[blockIdx.x] + sm[t] - v;    // exclusive within block + block offset
    if (i < n) { rowstart[i] = r; cursor[i] = r; }
    if (i == n - 1) rowstart[n] = E;
}

// fill CSR: csr_src[pos] = src, csr_w[pos] = dis[src]
__global__ __launch_bounds__(256) void csr_fill(const int* __restrict__ src,
                                                const int* __restrict__ dst,
                                                const float* __restrict__ dis,
                                                int* __restrict__ cursor,
                                                int* __restrict__ csr_src,
                                                float* __restrict__ csr_w, int e) {
    int i = blockIdx.x * 256 + threadIdx.x;
    if (i >= e) return;
    const int d = dst[i];
    const int s = src[i];
    const int pos = atomicAdd(&cursor[d], 1);
    csr_src[pos] = s;
    csr_w[pos] = dis[s];
}

// ================= WMMA dense matmul: Out[N,128] = A[N,K] @ W[K,128] =================
// Block = 256 thr (8 wave32): 64 rows x 128 cols (4 row-tiles of 16).  W staged once in
// LDS via gfx1250 async-to-LDS DMA and reused by all 4 tiles; A tile staged with fused
// concat/tanh transform at padded stride K+1 (odd => conflict-free LDS banks).  Each wave
// keeps 4 independent WMMA accumulator chains for ILP and 4x B-fragment reuse.

#define MODE_FEATCOND 0   // A = [feature | condition], K = 160
#define MODE_TANH     1   // A = tanh(buf[N,128]),      K = 128
#define MODE_ZCOND    2   // A = [z | condition],       K = 96

template<int K, int MODE, bool DUALB>
__global__ __launch_bounds__(256)
void mm_wmma(const float* __restrict__ A0, const float* __restrict__ A1,
             const float* __restrict__ W1, const float* __restrict__ W2,
             float* __restrict__ Out, int n)
{
    constexpr int KP = K + 1;            // padded LDS row stride (odd)
    __shared__ float As[64 * KP];
    __shared__ float Ws[K * 128];
    const int tid = threadIdx.x;
    const int rowbase = blockIdx.x * 64;

    // ---- stage W (B matrix) into LDS ----
    if (DUALB) {   // B = [W_mean | W_logvar], each [K,64]
        for (int t = tid; t < K * 128; t += 256) {
            const int r = t >> 7, c = t & 127;
            Ws[t] = (c < 64) ? W1[r * 64 + c] : W2[r * 64 + (c - 64)];
        }
    } else {
#if ASYNC_LDS_OK
        {
            auto gp = (__attribute__((address_space(1))) v4i_vs*)W1;   // AS1, non-const, v4i
            auto lp = (__attribute__((address_space(3))) v4i_vs*)Ws;   // AS3 v4i
            for (int t = tid; t < K * 32; t += 256)                    // K*32 16B vectors
                __builtin_amdgcn_global_load_async_to_lds_b128(gp + t, lp + t, 0, 0);
            __builtin_amdgcn_s_wait_asynccnt(0);
        }
#else
        for (int t = tid; t < K * 128; t += 256) Ws[t] = W1[t];
#endif
    }

    // ---- stage A tile (64 rows) with fused transform ----
    for (int t = tid; t < 64 * K; t += 256) {
        const int r = t / K;
        const int k = t - r * K;
        const int g = rowbase + r;
        float v = 0.0f;
        if (g < n) {
            if (MODE == MODE_FEATCOND) {
                v = (k < FEAT) ? A0[(size_t)g * FEAT + k] : A1[(size_t)g * COND + (k - FEAT)];
            } else if (MODE == MODE_TANH) {
                v = tanhf(A0[(size_t)g * HID + k]);
            } else { // MODE_ZCOND
                v = (k < LAT) ? A0[(size_t)g * LAT + k] : A1[(size_t)g * COND + (k - LAT)];
            }
        }
        As[r * KP + k] = v;
    }
    __syncthreads();

    const int warp = tid >> 5;
    const int lane = tid & 31;
    const int half = lane >> 4;      // 0: lanes 0-15, 1: lanes 16-31
    const int lr   = lane & 15;
    const int col  = warp * 16 + lr;

    v8f acc0 = {}, acc1 = {}, acc2 = {}, acc3 = {};
    #pragma unroll 2
    for (int k0 = 0; k0 < K; k0 += 4) {
        const int ka = k0 + 2 * half;           // 16x4 f32 A layout: lanes>=16 hold K+2,K+3
        v2f b;
        b.x = Ws[ka * 128 + col];
        b.y = Ws[(ka + 1) * 128 + col];
        v2f a0, a1, a2, a3;
        a0.x = As[(lr +  0) * KP + ka];  a0.y = As[(lr +  0) * KP + ka + 1];
        a1.x = As[(lr + 16) * KP + ka];  a1.y = As[(lr + 16) * KP + ka + 1];
        a2.x = As[(lr + 32) * KP + ka];  a2.y = As[(lr + 32) * KP + ka + 1];
        a3.x = As[(lr + 48) * KP + ka];  a3.y = As[(lr + 48) * KP + ka + 1];
        acc0 = __builtin_amdgcn_wmma_f32_16x16x4_f32(false, a0, false, b, (short)0, acc0, false, false);
        acc1 = __builtin_amdgcn_wmma_f32_16x16x4_f32(false, a1, false, b, (short)0, acc1, false, false);
        acc2 = __builtin_amdgcn_wmma_f32_16x16x4_f32(false, a2, false, b, (short)0, acc2, false, false);
        acc3 = __builtin_amdgcn_wmma_f32_16x16x4_f32(false, a3, false, b, (short)0, acc3, false, false);
    }

    // D-tile stores: uniform fast path per tile (no per-element exec juggling)
    auto store_tile = [&](const v8f& acc, int rb) {
        if (rb + 16 <= n) {
            #pragma unroll
            for (int v = 0; v < 8; v++)
                Out[(size_t)(rb + v + half * 8) * 128 + col] = acc[v];
        } else {
            #pragma unroll
            for (int v = 0; v < 8; v++) {
                const int row = rb + v + half * 8;
                if (row < n) Out[(size_t)row * 128 + col] = acc[v];
            }
        }
    };
    store_tile(acc0, rowbase +  0);
    store_tile(acc1, rowbase + 16);
    store_tile(acc2, rowbase + 32);
    store_tile(acc3, rowbase + 48);
}

// ================= GCN aggregation: CSR gather-reduce (no float atomics) =================
// One wave per node: 32 lanes x float4 = 128 features held in registers.
// O[i,:] = bias + H[i,:]*dis_i^2 + sum_{j in row i} H[csr_src[j],:] * (csr_w[j]*dis_i)

__global__ __launch_bounds__(256)
void conv_row(const float* __restrict__ H, const float* __restrict__ dis,
              const int* __restrict__ rowstart, const int* __restrict__ csr_src,
              const float* __restrict__ csr_w,
              const float* __restrict__ b1, const float* __restrict__ b2,
              float* __restrict__ O, int n)
{
    const int wid  = (blockIdx.x * 256 + threadIdx.x) >> 5;
    const int lane = threadIdx.x & 31;
    if (wid >= n) return;

    const float4* H4 = reinterpret_cast<const float4*>(H);
    const float di = dis[wid];

    float4 bias;
    {
        const int c = lane * 4;
        const float* bb = b1; int cc = c;
        if (b2 && c >= 64) { bb = b2; cc = c - 64; }
        bias = {bb[cc], bb[cc + 1], bb[cc + 2], bb[cc + 3]};
    }

    const float self = di * di;           // self-loop norm = 1/deg
    float4 acc = H4[(size_t)wid * 32 + lane];
    acc.x = bias.x + acc.x * self;
    acc.y = bias.y + acc.y * self;
    acc.z = bias.z + acc.z * self;
    acc.w = bias.w + acc.w * self;

    const int jend = rowstart[wid + 1];
    for (int j = rowstart[wid]; j < jend; ++j) {
        const int s   = csr_src[j];
        const float w = csr_w[j] * di;
        const float4 h = H4[(size_t)s * 32 + lane];
        acc.x += h.x * w;
        acc.y += h.y * w;
        acc.z += h.z * w;
        acc.w += h.w * w;
    }
    reinterpret_cast<float4*>(O)[(size_t)wid * 32 + lane] = acc;
}

// ================= reparameterization + output split =================

__global__ __launch_bounds__(256)
void z_split(const float* __restrict__ ML, const float* __restrict__ noise,
             float* __restrict__ zO, float* __restrict__ meanO,
             float* __restrict__ logvO, int n)
{
    int idx = blockIdx.x * 256 + threadIdx.x;
    if (idx >= n * LAT) return;
    const int i = idx / LAT;
    const int c = idx - i * LAT;
    const float mean = ML[(size_t)i * 128 + c];
    const float logv = ML[(size_t)i * 128 + 64 + c];
    const float z = noise[idx] * __expf(0.5f * logv) + mean;
    zO[idx]    = z;
    meanO[idx] = mean;
    logvO[idx] = logv;
}

// ================= driver =================

extern "C" void kernel_launch(void* const* d_in, const int* in_sizes, int n_in,
                              void* d_out, int out_size, void* d_ws, size_t ws_size,
                              hipStream_t stream)
{
    const float* feature      = (const float*)d_in[0];
    const float* condition    = (const float*)d_in[1];
    const float* noise        = (const float*)d_in[2];
    const int*   edge         = (const int*)d_in[3];
    const float* enc_i2h_W    = (const float*)d_in[4];
    const float* enc_i2h_b    = (const float*)d_in[5];
    const float* enc_h2h_W    = (const float*)d_in[6];
    const float* enc_h2h_b    = (const float*)d_in[7];
    const float* enc_mean_W   = (const float*)d_in[8];
    const float* enc_mean_b   = (const float*)d_in[9];
    const float* enc_logvar_W = (const float*)d_in[10];
    const float* enc_logvar_b = (const float*)d_in[11];
    const float* dec_i2h_W    = (const float*)d_in[12];
    const float* dec_i2h_b    = (const float*)d_in[13];
    const float* dec_h2h_W    = (const float*)d_in[14];
    const float* dec_h2h_b    = (const float*)d_in[15];
    const float* dec_out_W    = (const float*)d_in[16];
    const float* dec_out_b    = (const float*)d_in[17];

    const int n = in_sizes[0] / FEAT;
    const int E = in_sizes[3] / 2;
    const int* src = edge;
    const int* dst = edge + E;

    // workspace layout
    const size_t sn = (size_t)n;
    float* MM       = (float*)d_ws;              // n*128
    float* CV       = MM + sn * 128;             // n*128
    float* dis      = CV + sn * 128;             // n
    int*   cnt      = (int*)(dis + sn);          // n
    int*   rowstart = cnt + sn;                  // n+1
    int*   cursor   = rowstart + (sn + 1);       // n
    int*   bsum     = cursor + sn;               // <=1024
    int*   csr_src  = bsum + 1024;               // E
    float* csr_w    = (float*)(csr_src + (size_t)E); // E

    float* out_z    = (float*)d_out;             // [n,64]
    float* out_mean = out_z    + sn * LAT;
    float* out_logv = out_mean + sn * LAT;
    float* out_dec  = out_logv + sn * LAT;       // [n,128]

    const int nB  = (n + 255) / 256;
    const int eB  = (E + 255) / 256;
    const int mmB = (n + 63) / 64;               // 64 rows per block
    const int cvB = (n + 7) / 8;                 // 8 waves/block, 1 node/wave
    const int zB  = (n * LAT + 255) / 256;
    const int nbS = (n + 255) / 256;             // #scan partials (<=1024)

    // ---- CSR + normalization build ----
    zero_i32       <<<nB, 256, 0, stream>>>(cnt, n);
    count_dst      <<<eB, 256, 0, stream>>>(dst, cnt, E);
    dis_k          <<<nB, 256, 0, stream>>>(cnt, dis, n);
    scan_block_sums<<<nbS, 256, 0, stream>>>(cnt, bsum, n);
    scan_partials  <<<1, 1024, 0, stream>>>(bsum, nbS);
    scan_final     <<<nbS, 256, 0, stream>>>(cnt, bsum, rowstart, cursor, n, E);
    csr_fill       <<<eB, 256, 0, stream>>>(src, dst, dis, cursor, csr_src, csr_w, E);

    // ---- encoder conv1: x = [feature|condition] ----
    mm_wmma<160, MODE_FEATCOND, false><<<mmB, 256, 0, stream>>>(feature, condition, enc_i2h_W, nullptr, MM, n);
    conv_row<<<cvB, 256, 0, stream>>>(MM, dis, rowstart, csr_src, csr_w, enc_i2h_b, nullptr, CV, n);

    // ---- encoder conv2 (tanh fused into A load) ----
    mm_wmma<128, MODE_TANH, false><<<mmB, 256, 0, stream>>>(CV, nullptr, enc_h2h_W, nullptr, MM, n);
    conv_row<<<cvB, 256, 0, stream>>>(MM, dis, rowstart, csr_src, csr_w, enc_h2h_b, nullptr, CV, n);

    // ---- mean|logvar fused conv ----
    mm_wmma<128, MODE_TANH, true><<<mmB, 256, 0, stream>>>(CV, nullptr, enc_mean_W, enc_logvar_W, MM, n);
    conv_row<<<cvB, 256, 0, stream>>>(MM, dis, rowstart, csr_src, csr_w, enc_mean_b, enc_logvar_b, CV, n);

    // ---- reparameterize + emit z/mean/logvar ----
    z_split<<<zB, 256, 0, stream>>>(CV, noise, out_z, out_mean, out_logv, n);

    // ---- decoder conv1: x2 = [z|condition] ----
    mm_wmma<96, MODE_ZCOND, false><<<mmB, 256, 0, stream>>>(out_z, condition, dec_i2h_W, nullptr, MM, n);
    conv_row<<<cvB, 256, 0, stream>>>(MM, dis, rowstart, csr_src, csr_w, dec_i2h_b, nullptr, CV, n);

    // ---- decoder conv2 ----
    mm_wmma<128, MODE_TANH, false><<<mmB, 256, 0, stream>>>(CV, nullptr, dec_h2h_W, nullptr, MM, n);
    conv_row<<<cvB, 256, 0, stream>>>(MM, dis, rowstart, csr_src, csr_w, dec_h2h_b, nullptr, CV, n);

    // ---- decoder output conv -> d_out ----
    mm_wmma<128, MODE_TANH, false><<<mmB, 256, 0, stream>>>(CV, nullptr, dec_out_W, nullptr, MM, n);
    conv_row<<<cvB, 256, 0, stream>>>(MM, dis, rowstart, csr_src, csr_w, dec_out_b, nullptr, out_dec, n);
}